// LNSNet_90623809945930
// MI455X (gfx1250) — compile-verified
//
#include <hip/hip_runtime.h>
#include <math.h>

typedef float v2f __attribute__((ext_vector_type(2)));
typedef float v8f __attribute__((ext_vector_type(8)));

// ---------------------------------------------------------------------------
// Kernel 1: conv1 (1->32, 3x3, valid) + bias + ReLU + 2x2 maxpool, fused.
// x: [512,1,54,54] -> pool1: [512,32,26,26]
// Memory-bound (9 MACs/tap); one thread per pooled output.
// ---------------------------------------------------------------------------
__global__ void __launch_bounds__(256)
k_conv1_relu_pool(const float* __restrict__ x, const float* __restrict__ w,
                  const float* __restrict__ bias, float* __restrict__ pool1) {
  const int TOTAL = 512 * 32 * 26 * 26;
  int idx = blockIdx.x * 256 + threadIdx.x;
  if (idx >= TOTAL) return;
  int px = idx % 26; int t = idx / 26;
  int py = t % 26;   t /= 26;
  int c  = t % 32;   int b = t / 32;
  const float* img = x + b * (54 * 54);
  const float* wc  = w + c * 9;
  float w00 = wc[0], w01 = wc[1], w02 = wc[2];
  float w10 = wc[3], w11 = wc[4], w12 = wc[5];
  float w20 = wc[6], w21 = wc[7], w22 = wc[8];
  float bn = bias[c];
  float best = 0.0f;                 // ReLU outputs >= 0 -> 0 is the max identity
#pragma unroll
  for (int dy = 0; dy < 2; ++dy)
#pragma unroll
    for (int dx = 0; dx < 2; ++dx) {
      int oy = 2 * py + dy, ox = 2 * px + dx;
      const float* p = img + oy * 54 + ox;
      float s = bn
              + p[0]   * w00 + p[1]   * w01 + p[2]   * w02
              + p[54]  * w10 + p[55]  * w11 + p[56]  * w12
              + p[108] * w20 + p[109] * w21 + p[110] * w22;
      best = fmaxf(best, fmaxf(s, 0.0f));
    }
  pool1[idx] = best;
}

// ---------------------------------------------------------------------------
// Kernel 2: conv2 (32->64, 3x3) as implicit GEMM on V_WMMA_F32_16X16X4_F32.
// GEMM: M = 512*24*24 (row order m = ((b*12+py)*12+px)*4 + pool-slot),
//       N = 64 channels, K = 32*9 = 288.
// Workgroup = 128 threads (4 waves) covers a 32(M) x 64(N) macro-tile:
//   - 32x288 im2col A staged in LDS (289-float stride: conflict-free),
//   - each wave owns 16 N-columns and TWO accumulators (M blocking x2):
//     both WMMAs share the B fragment -> half the B traffic per FLOP and two
//     hazard-free back-to-back wmma issues.
// Epilogue fuses bias + ReLU + 2x2 maxpool in registers (each pool window is
// 4 consecutive C rows = 4 consecutive acc VGPRs) and writes flattened
// h2 [512, 9216] (c*144 + y*12 + x).
// ---------------------------------------------------------------------------
__global__ void __launch_bounds__(128)
k_conv2_wmma(const float* __restrict__ pool1, const float* __restrict__ w2,
             const float* __restrict__ b2, float* __restrict__ h2) {
  __shared__ float As[32][289];
  const int tid  = threadIdx.x;
  const int tile = blockIdx.x;                // 0 .. 9215 (M macro-tiles of 32)

  // Cooperative im2col gather: 32 rows x 288 K (9216 = 128*72, uniform).
  for (int i = tid; i < 32 * 288; i += 128) {
    int row = i / 288, k = i - row * 288;
    int m = tile * 32 + row;
    int q = m & 3, s = m >> 2;                // q = slot in 2x2 pool window
    int dx = q & 1, dy = q >> 1;
    int pxp = s % 12; s /= 12;
    int pyp = s % 12; int b = s / 12;
    int oy = 2 * pyp + dy, ox = 2 * pxp + dx; // conv2 output coords (24x24)
    int cc = k / 9; int r = k - cc * 9;
    int ky = r / 3, kx = r - ky * 3;
    As[row][k] = pool1[((b * 32 + cc) * 26 + (oy + ky)) * 26 + (ox + kx)];
  }
  __syncthreads();

  const int lane = tid & 31;
  const int wave = tid >> 5;
  const int hi = lane >> 4, lo = lane & 15;
  const int n = wave * 16 + lo;               // output channel 0..63
  const float* bw = w2 + n * 288;             // B[k][n] = conv2_w[n*288 + k]

  v8f acc0 = {}, acc1 = {};
#pragma unroll 4
  for (int ks = 0; ks < 72; ++ks) {           // K = 288, 4 per WMMA
    int kk = ks * 4 + hi * 2;
    v2f a0; a0.x = As[lo][kk];      a0.y = As[lo][kk + 1];
    v2f a1; a1.x = As[lo + 16][kk]; a1.y = As[lo + 16][kk + 1];
    v2f bb = *(const v2f*)(bw + kk);          // 8B aligned (kk even, row even)
    acc0 = __builtin_amdgcn_wmma_f32_16x16x4_f32(false, a0, false, bb,
                                                 (short)0, acc0, false, false);
    acc1 = __builtin_amdgcn_wmma_f32_16x16x4_f32(false, a1, false, bb,
                                                 (short)0, acc1, false, false);
  }

  // Epilogue: bias + ReLU + 2x2 maxpool per accumulator.
  float bn = b2[n];
#pragma unroll
  for (int half = 0; half < 2; ++half) {
    v8f acc = half ? acc1 : acc0;
    float r0 = fmaxf(acc[0] + bn, 0.f), r1 = fmaxf(acc[1] + bn, 0.f);
    float r2 = fmaxf(acc[2] + bn, 0.f), r3 = fmaxf(acc[3] + bn, 0.f);
    float r4 = fmaxf(acc[4] + bn, 0.f), r5 = fmaxf(acc[5] + bn, 0.f);
    float r6 = fmaxf(acc[6] + bn, 0.f), r7 = fmaxf(acc[7] + bn, 0.f);
    float pa = fmaxf(fmaxf(r0, r1), fmaxf(r2, r3));
    float pb = fmaxf(fmaxf(r4, r5), fmaxf(r6, r7));
    int p0 = tile * 8 + half * 4 + hi * 2;    // pooled index = (b*12+py)*12+px
    {
      int px = p0 % 12, t = p0 / 12; int py = t % 12; int b = t / 12;
      h2[b * 9216 + n * 144 + py * 12 + px] = pa;
    }
    int p1 = p0 + 1;
    {
      int px = p1 % 12, t = p1 / 12; int py = t % 12; int b = t / 12;
      h2[b * 9216 + n * 144 + py * 12 + px] = pb;
    }
  }
}

// ---------------------------------------------------------------------------
// Kernel 3: fc1 GEMM on WMMA f32. M=512, N=128, K=9216.
// a1[m][n] = relu( sum_k h2[m][k] * fc1_w[n][k] + fc1_b[n] )
// Single-wave workgroups (no LDS/barrier needed) -> 256 independent WGs for
// occupancy. Aligned 8-byte fragment loads; fc1_w (4.7 MB) is L2-resident and
// stride-prefetched via global_prefetch_b8.
// ---------------------------------------------------------------------------
__global__ void __launch_bounds__(32)
k_fc1_wmma(const float* __restrict__ h2, const float* __restrict__ fw,
           const float* __restrict__ fb, float* __restrict__ a1) {
  const int lane = threadIdx.x;                       // one wave per WG
  const int hi = lane >> 4, lo = lane & 15;
  const int m0 = blockIdx.x * 16;                     // gridDim.x = 32
  const int n  = blockIdx.y * 16 + lo;                // gridDim.y = 8
  const float* arow = h2 + (m0 + lo) * 9216;
  const float* brow = fw + n * 9216;

  v8f acc = {};
#pragma unroll 4
  for (int ks = 0; ks < 2304; ++ks) {                 // K = 9216, 4 per WMMA
    int kk = ks * 4 + hi * 2;
    if ((ks & 31) == 0) {                             // uniform branch: EXEC intact
      __builtin_prefetch(arow + kk + 256, 0, 0);
      __builtin_prefetch(brow + kk + 256, 0, 0);
    }
    v2f a  = *(const v2f*)(arow + kk);
    v2f bb = *(const v2f*)(brow + kk);
    acc = __builtin_amdgcn_wmma_f32_16x16x4_f32(false, a, false, bb,
                                                (short)0, acc, false, false);
  }
  float bn = fb[n];
#pragma unroll
  for (int v = 0; v < 8; ++v) {
    int m = m0 + v + 8 * hi;
    a1[m * 128 + n] = fmaxf(acc[v] + bn, 0.f);
  }
}

// ---------------------------------------------------------------------------
// Kernel 4: fc2 (128->10), tiny: plain VALU dot products.
// ---------------------------------------------------------------------------
__global__ void __launch_bounds__(256)
k_fc2(const float* __restrict__ a1, const float* __restrict__ fw,
      const float* __restrict__ fb, float* __restrict__ out) {
  int idx = blockIdx.x * 256 + threadIdx.x;           // 512*10 = 5120
  if (idx >= 512 * 10) return;
  int nn = idx % 10, m = idx / 10;
  const float* ar = a1 + m * 128;
  const float* wr = fw + nn * 128;
  float s = fb[nn];
#pragma unroll 8
  for (int k = 0; k < 128; ++k) s += ar[k] * wr[k];
  out[idx] = s;
}

// ---------------------------------------------------------------------------
extern "C" void kernel_launch(void* const* d_in, const int* in_sizes, int n_in,
                              void* d_out, int out_size, void* d_ws, size_t ws_size,
                              hipStream_t stream) {
  (void)in_sizes; (void)n_in; (void)out_size; (void)ws_size;
  const float* x   = (const float*)d_in[0];
  const float* c1w = (const float*)d_in[1];
  const float* c1b = (const float*)d_in[2];
  const float* c2w = (const float*)d_in[3];
  const float* c2b = (const float*)d_in[4];
  const float* f1w = (const float*)d_in[5];
  const float* f1b = (const float*)d_in[6];
  const float* f2w = (const float*)d_in[7];
  const float* f2b = (const float*)d_in[8];
  float* out = (float*)d_out;

  // Workspace layout (total ~63.5 MB):
  //   pool1 [512,32,26,26] f32 : 44,335,104 B
  //   h2    [512,9216]     f32 : 18,874,368 B
  //   a1    [512,128]      f32 :    262,144 B
  char* ws = (char*)d_ws;
  float* pool1 = (float*)(ws);
  float* h2    = (float*)(ws + 44335104);
  float* a1    = (float*)(ws + 44335104 + 18874368);

  const int POOL1_TOTAL = 512 * 32 * 26 * 26;
  k_conv1_relu_pool<<<(POOL1_TOTAL + 255) / 256, 256, 0, stream>>>(x, c1w, c1b, pool1);
  k_conv2_wmma<<<512 * 24 * 24 / 32, 128, 0, stream>>>(pool1, c2w, c2b, h2);
  dim3 g1(32, 8);
  k_fc1_wmma<<<g1, 32, 0, stream>>>(h2, f1w, f1b, a1);
  k_fc2<<<(5120 + 255) / 256, 256, 0, stream>>>(a1, f2w, f2b, out);
}